// MEModel_23562190586224
// MI455X (gfx1250) — compile-verified
//
#include <hip/hip_runtime.h>
#include <stdint.h>

// Sparse 3x3x3 conv (C_in=C_out=1) via L2-resident padded dense grid.
//   G=256, padded extent GP=258 -> grid = 258^3 floats = 68.7 MB (< 192 MB L2).
//   Phase 1: zero grid. Phase 2: scatter feats. Phase 3: 27-tap gather-FMA.
// Streams use NT hints to preserve L2 residency of the grid.
// Weights are staged into LDS via the CDNA5 async-to-LDS path (ASYNCcnt).

#define G      256
#define GP     258
#define PLANE  (GP * GP)                    // 66564
#define GRID_ELEMS ((size_t)GP * GP * GP)   // 17,173,512 floats = 68.7 MB

typedef __attribute__((address_space(3))) float lds_f32;

__global__ void zero_grid_kernel(float4* __restrict__ g, int n4) {
    int i = blockIdx.x * blockDim.x + threadIdx.x;
    if (i < n4) {
        float4 z = {0.0f, 0.0f, 0.0f, 0.0f};
        g[i] = z;   // default RT store: lines land (clean-ish) in L2
    }
}

__global__ void scatter_kernel(const int* __restrict__ coords,
                               const float* __restrict__ feats,
                               float* __restrict__ grid, int n) {
    int i = blockIdx.x * blockDim.x + threadIdx.x;
    if (i >= n) return;
    // Streaming reads: NT so they don't displace the grid in L2.
    int x = __builtin_nontemporal_load(coords + 3 * i + 0);
    int y = __builtin_nontemporal_load(coords + 3 * i + 1);
    int z = __builtin_nontemporal_load(coords + 3 * i + 2);
    float f = __builtin_nontemporal_load(feats + i);
    int lin = ((x + 1) * GP + (y + 1)) * GP + (z + 1);
    grid[lin] = f;  // RT store: keep grid resident in L2 for the gather phase
}

__global__ void gather27_kernel(const int* __restrict__ coords,
                                const float* __restrict__ grid,
                                const float* __restrict__ W,
                                float* __restrict__ out, int n) {
    __shared__ float sW[32];

    // --- CDNA5 async-to-LDS broadcast of the 27 weights (wave 0 lanes 0..26).
    // Per-lane copy: global dword at SGPR base + VGPR offset goes to the LDS address in vdst; tracked by ASYNCcnt.
    int tid = threadIdx.x;
    if (tid < 27) {
        lds_f32* sp = (lds_f32*)&sW[0];
        unsigned ldsaddr = (unsigned)(size_t)(sp + tid);  // raw LDS byte address
        unsigned goff    = (unsigned)tid * 4u;            // byte offset vs SGPR base
        asm volatile("global_load_async_to_lds_b32 %0, %1, %2"
                     :: "v"(ldsaddr), "v"(goff), "s"(W)
                     : "memory");
    }
    asm volatile("s_wait_asynccnt 0" ::: "memory");
    __syncthreads();

    int i = blockIdx.x * blockDim.x + tid;
    if (i >= n) return;

    int x = __builtin_nontemporal_load(coords + 3 * i + 0);
    int y = __builtin_nontemporal_load(coords + 3 * i + 1);
    int z = __builtin_nontemporal_load(coords + 3 * i + 2);
    int base = ((x + 1) * GP + (y + 1)) * GP + (z + 1);

    // 27-tap stencil: all taps valid thanks to the zero pad ring; absent
    // neighbors read 0.0f, exactly matching the reference's `where(found,...,0)`.
    float acc = 0.0f;
    const float* g0 = grid + base;
#pragma unroll
    for (int dx = 0; dx < 3; ++dx) {
#pragma unroll
        for (int dy = 0; dy < 3; ++dy) {
            const float* row = g0 + (dx - 1) * PLANE + (dy - 1) * GP - 1;
#pragma unroll
            for (int dz = 0; dz < 3; ++dz) {
                acc = fmaf(row[dz], sW[dx * 9 + dy * 3 + dz], acc);
            }
        }
    }
    __builtin_nontemporal_store(acc, out + i);  // streaming output
}

extern "C" void kernel_launch(void* const* d_in, const int* in_sizes, int n_in,
                              void* d_out, int out_size, void* d_ws, size_t ws_size,
                              hipStream_t stream) {
    const int*   coords = (const int*)d_in[0];   // (N,3) int32
    const float* feats  = (const float*)d_in[1]; // (N,1) f32
    const float* W      = (const float*)d_in[2]; // (27,1,1) f32
    float* out  = (float*)d_out;                 // (N,1) f32
    float* grid = (float*)d_ws;                  // needs >= 68.7 MB scratch

    int n = in_sizes[1];                         // N = 2,000,000
    if (n <= 0) return;

    const int n4 = (int)(GRID_ELEMS / 4);        // 4,293,378 float4 stores
    zero_grid_kernel<<<(n4 + 255) / 256, 256, 0, stream>>>((float4*)grid, n4);
    scatter_kernel<<<(n + 255) / 256, 256, 0, stream>>>(coords, feats, grid, n);
    gather27_kernel<<<(n + 255) / 256, 256, 0, stream>>>(coords, grid, W, out, n);
}